// Part_Attention_clip_67542655697210
// MI455X (gfx1250) — compile-verified
//
#include <hip/hip_runtime.h>

typedef __attribute__((ext_vector_type(2))) float v2f;
typedef __attribute__((ext_vector_type(8))) float v8f;

#define BATCH    128
#define N201     201
#define LDSTRIDE 209   // odd stride -> conflict-free LDS column gathers
#define NWAVES   13
#define NTHREADS (NWAVES * 32)

__device__ __forceinline__ int pmap(int t, int e, int lo) {
    // att index -> original 201-grid index: 0 -> e, t>=1 -> lo+t-1
    return (t == 0) ? e : (lo + t - 1);
}

__global__ __launch_bounds__(NTHREADS)
void part_att_chain_kernel(const float* __restrict__ x,
                           float* __restrict__ out_f,
                           int*   __restrict__ out_i)
{
    __shared__ float sbuf[2][16 * LDSTRIDE];

    const int e = blockIdx.x;   // expert 0..4  (adjacent blocks share layer windows in L2)
    const int b = blockIdx.y;   // batch  0..127

    const int lo_tab[5] = {5, 5, 5, 54, 103};
    const int hi_tab[5] = {201, 201, 103, 152, 201};
    const int lo = lo_tab[e], hi = hi_tab[e];
    const int M  = 1 + hi - lo;            // 197 or 99
    const int Mp = (M + 15) & ~15;         // 208 or 112
    const int NT = Mp >> 4;                // 16-col tiles: 13 or 7
    const int KT = Mp >> 2;                // K chunks of 4: 52 or 28

    const int tid  = threadIdx.x;
    const int lane = tid & 31;
    const int wid  = tid >> 5;
    const int lp   = lane & 15;
    const int half = lane >> 4;

    const size_t mat = (size_t)N201 * N201;

    // ---- init R = att[11][0:16, 0:M], zero-padded to Mp columns
    {
        const float* __restrict__ X11 = x + ((size_t)(11 * BATCH + b)) * mat;
        for (int idx = tid; idx < 16 * Mp; idx += NTHREADS) {
            int m = idx / Mp, n = idx - m * Mp;
            float v = 0.f;
            if (n < M)
                v = X11[(size_t)pmap(m, e, lo) * N201 + pmap(n, e, lo)];
            sbuf[0][m * LDSTRIDE + n] = v;
        }
    }
    __syncthreads();

    int cur = 0;
    for (int s = 0; s < 11; ++s) {
        const int l = 10 - s;                                  // att[10] .. att[0]
        const float* __restrict__ Xl = x + ((size_t)(l * BATCH + b)) * mat;
        const float* __restrict__ src = sbuf[cur];
        float*       __restrict__ dst = sbuf[cur ^ 1];

        for (int jt = wid; jt < NT; jt += NWAVES) {            // wave-uniform -> EXEC all-1s
            const int n  = jt * 16 + lp;                       // output column (padded space)
            const int cn = pmap(n < M ? n : (M - 1), e, lo);   // clamped global column
            v8f acc = {};
            for (int kt = 0; kt < KT; ++kt) {
                const int k0 = kt * 4 + 2 * half;
                // A-fragment (16x4) from running block R in LDS: m=lp, k=k0+v
                v2f a;
                a.x = src[lp * LDSTRIDE + k0];
                a.y = src[lp * LDSTRIDE + k0 + 1];
                // B-fragment (4x16) straight from global layer matrix: k=k0+v, n fixed
                const int r0 = pmap(k0     < M ? k0     : (M - 1), e, lo);
                const int r1 = pmap(k0 + 1 < M ? k0 + 1 : (M - 1), e, lo);
                v2f bb;
                bb.x = Xl[(size_t)r0 * N201 + cn];
                bb.y = Xl[(size_t)r1 * N201 + cn];
                acc = __builtin_amdgcn_wmma_f32_16x16x4_f32(
                    /*neg_a=*/false, a, /*neg_b=*/false, bb,
                    /*c_mod=*/(short)0, acc, /*reuse_a=*/false, /*reuse_b=*/false);
            }
            // D (16x16): d[v] -> row (v + 8*half), col n. Keep padded cols exactly 0.
            #pragma unroll
            for (int v = 0; v < 8; ++v) {
                float val = (n < M) ? acc[v] : 0.f;
                dst[(v + 8 * half) * LDSTRIDE + n] = val;
            }
        }
        __syncthreads();
        cur ^= 1;
    }

    // ---- row 0, cols 1..M-1: max + first-occurrence argmax
    if (tid == 0) {
        const float* row = sbuf[cur];
        float best = row[1];
        int   bj   = 1;
        for (int j = 2; j < M; ++j) {
            float v = row[j];
            if (v > best) { best = v; bj = j; }
        }
        out_f[e * BATCH + b] = best;                       // max_p  [5,128,1] f32
        out_i[5 * BATCH + e * BATCH + b] = lo + bj - 1;    // max_inx[5,128,1] i32 (raw bits)
    }
}

extern "C" void kernel_launch(void* const* d_in, const int* in_sizes, int n_in,
                              void* d_out, int out_size, void* d_ws, size_t ws_size,
                              hipStream_t stream) {
    const float* x = (const float*)d_in[0];
    float* out_f = (float*)d_out;
    int*   out_i = (int*)d_out;   // second half holds int32 bit patterns
    dim3 grid(5, BATCH);          // experts adjacent -> shared windows hit in L2
    part_att_chain_kernel<<<grid, NTHREADS, 0, stream>>>(x, out_f, out_i);
}